// TorchRgcnLayer_188978561158
// MI455X (gfx1250) — compile-verified
//
#include <hip/hip_runtime.h>

#define N_ENT   50000
#define NUM_REL 41
#define DIM     64
#define N_EDGES 1600000

typedef float v2f __attribute__((ext_vector_type(2)));
typedef float v4f __attribute__((ext_vector_type(4)));
typedef float v8f __attribute__((ext_vector_type(8)));

typedef unsigned int u32x4 __attribute__((ext_vector_type(4)));
typedef int i32x4 __attribute__((ext_vector_type(4)));
typedef int i32x8 __attribute__((ext_vector_type(8)));

// ---------------------------------------------------------------------------
// Zero the output accumulator and the degree histogram.
__global__ void rgcn_zero_kernel(float* __restrict__ out, unsigned* __restrict__ deg,
                                 int n_out, int n_deg) {
    int i = blockIdx.x * blockDim.x + threadIdx.x;
    int stride = gridDim.x * blockDim.x;
    for (int k = i; k < n_out; k += stride) out[k] = 0.0f;
    for (int k = i; k < n_deg; k += stride) deg[k] = 0u;
}

// ---------------------------------------------------------------------------
// deg[et*N + fr] += 1 for every real edge (self-loops have degree 1 by
// construction and are handled with val = 1.0 in the finalize kernel).
__global__ void rgcn_degree_kernel(const int* __restrict__ fr,
                                   const int* __restrict__ et,
                                   unsigned* __restrict__ deg) {
    int i = blockIdx.x * blockDim.x + threadIdx.x;
    int stride = gridDim.x * blockDim.x;
    for (int e = i; e < N_EDGES; e += stride) {
        // Pull the streaming edge lists ahead (global_prefetch_b8).
        if (e + stride < N_EDGES) {
            __builtin_prefetch(&et[e + stride], 0, 1);
            __builtin_prefetch(&fr[e + stride], 0, 1);
        }
        atomicAdd(&deg[et[e] * N_ENT + fr[e]], 1u);
    }
}

// ---------------------------------------------------------------------------
// Y[r, n, :] = X[n, :] @ W[r]   (fp32, V_WMMA_F32_16X16X4_F32)
// Block = 256 threads = 8 waves. Each block: one relation r (blockIdx.y) and
// 32 rows of X (blockIdx.x). Waves tile the 32x64 output as 2x4 16x16 tiles.
// W_r (16 KB) is DMA'd into LDS by the Tensor Data Mover (tensor_load_to_lds),
// issued once by wave 0 and tracked with TENSORcnt.
__global__ __launch_bounds__(256) void rgcn_gemm_wmma(
    const float* __restrict__ x, const float* __restrict__ w,
    float* __restrict__ y) {
    __shared__ float Wlds[DIM * DIM];

    const int r = blockIdx.y;
    const float* wr = w + (size_t)r * DIM * DIM;

    if (threadIdx.x == 0) {
        // Tensor DMA descriptor (CDNA5 ISA):
        // 1-D tile of 4096 fp32 elements, global -> LDS.
        unsigned long long ga = (unsigned long long)(uintptr_t)wr;
        unsigned lds = (unsigned)(uintptr_t)&Wlds[0];   // low 32 bits = LDS byte addr

        u32x4 g0;
        g0[0] = 1u;                                   // count=1, user descriptor
        g0[1] = lds;                                  // lds_addr
        g0[2] = (unsigned)(ga & 0xFFFFFFFFu);         // global_addr[31:0]
        g0[3] = (unsigned)((ga >> 32) & 0x1FFFFFFu)   // global_addr[56:32]
              | (2u << 30);                           // type=2 ("image")

        i32x8 g1;
        g1[0] = (2 << 16);                 // workgroup_mask=0, data_size=2 (4 bytes)
        g1[1] = (int)(4096u << 16);        // tensor_dim0[15:0]=4096 in [31:16]
        g1[2] = (1 << 16);                 // tensor_dim0[31:16]=0; tensor_dim1=1
        g1[3] = (int)(4096u << 16);        // tensor_dim1[31:16]=0; tile_dim0=4096
        g1[4] = 1;                         // tile_dim1=1, tile_dim2=0
        g1[5] = 4096;                      // tensor_dim0_stride[31:0] = 4096
        g1[6] = 0;                         // stride[47:32]=0; tensor_dim1_stride lo=0
        g1[7] = 0;

        i32x4 gz = {0, 0, 0, 0};           // groups 2/3 unused (<=2D tensor)
        i32x8 gz8 = {0, 0, 0, 0, 0, 0, 0, 0};
        __builtin_amdgcn_tensor_load_to_lds(g0, g1, gz, gz, gz8, 0);
        __builtin_amdgcn_s_wait_tensorcnt(0);
    }
    __syncthreads();

    const int wave = threadIdx.x >> 5;   // 0..7
    const int lane = threadIdx.x & 31;   // wave32
    const int m    = lane & 15;          // M index within tile
    const int kh   = lane >> 4;          // K half-select (0/1)
    const int rowBase = blockIdx.x * 32 + (wave >> 2) * 16;
    const int n0      = (wave & 3) * 16;
    const int col     = n0 + m;          // D/N column for this lane

    // A tile: 16 rows x 64 K -> 16 v2f per lane.
    // Lane layout per step s: VGPR0/1 = X[row, 4s + 2*kh + {0,1}].
    int arow = rowBase + m;
    if (arow >= N_ENT) arow = N_ENT - 1;   // clamp; EXEC must stay all-ones for WMMA
    const float* xr = x + (size_t)arow * DIM;
    v2f a[16];
#pragma unroll
    for (int s = 0; s < 16; ++s)
        a[s] = *(const v2f*)(xr + 4 * s + 2 * kh);

    v8f c = {};
#pragma unroll
    for (int s = 0; s < 16; ++s) {
        // B 4x16: VGPR0 = rows {4s, 4s+1} across lane halves, VGPR1 = {4s+2, 4s+3}.
        v2f b;
        b.x = Wlds[(4 * s + kh) * DIM + col];
        b.y = Wlds[(4 * s + 2 + kh) * DIM + col];
        c = __builtin_amdgcn_wmma_f32_16x16x4_f32(
            /*neg_a=*/false, a[s], /*neg_b=*/false, b,
            /*c_mod=*/(short)0, c, /*reuse_a=*/false, /*reuse_b=*/false);
    }

    // D 16x16 f32: VGPR j -> row j (lanes 0-15) / row j+8 (lanes 16-31).
    float* yb = y + ((size_t)r * N_ENT) * DIM;
#pragma unroll
    for (int j = 0; j < 8; ++j) {
        int row = rowBase + j + 8 * kh;
        if (row < N_ENT) yb[(size_t)row * DIM + col] = c[j];
    }
}

// ---------------------------------------------------------------------------
// out[fr, :] += (1/deg[et*N+fr]) * Y[et, to, :]   -- 16 lanes per edge,
// b128 gather from Y, fp32 atomics into L2-resident out.
__global__ __launch_bounds__(256) void rgcn_scatter_kernel(
    const int* __restrict__ fr, const int* __restrict__ to,
    const int* __restrict__ et, const unsigned* __restrict__ deg,
    const float* __restrict__ y, float* __restrict__ out) {
    const int lane16 = threadIdx.x & 15;
    const int e = blockIdx.x * 16 + (threadIdx.x >> 4);
    if (e >= N_EDGES) return;
    const int r = et[e];
    const int s = fr[e];
    const int t = to[e];
    const float val = 1.0f / (float)deg[r * N_ENT + s];
    const v4f yv = *(const v4f*)(y + ((size_t)r * N_ENT + t) * DIM + lane16 * 4);
    float* o = out + (size_t)s * DIM + lane16 * 4;
    atomicAdd(o + 0, val * yv.x);
    atomicAdd(o + 1, val * yv.y);
    atomicAdd(o + 2, val * yv.z);
    atomicAdd(o + 3, val * yv.w);
}

// ---------------------------------------------------------------------------
// Self-loop relation (R-1) has val = 1.0: out += Y[R-1] + bias.
// Also pass through the `r` tensor into the tail of d_out.
__global__ void rgcn_finalize_kernel(const float* __restrict__ y,
                                     const float* __restrict__ bias,
                                     const float* __restrict__ rvec,
                                     float* __restrict__ out) {
    const size_t total = (size_t)N_ENT * DIM;
    const float* yself = y + (size_t)(NUM_REL - 1) * N_ENT * DIM;
    size_t i = (size_t)blockIdx.x * blockDim.x + threadIdx.x;
    size_t stride = (size_t)gridDim.x * blockDim.x;
    for (size_t k = i; k < total; k += stride)
        out[k] += yself[k] + bias[k & (DIM - 1)];
    for (size_t k = i; k < (size_t)(NUM_REL - 1) * DIM; k += stride)
        out[total + k] = rvec[k];
}

// ---------------------------------------------------------------------------
extern "C" void kernel_launch(void* const* d_in, const int* in_sizes, int n_in,
                              void* d_out, int out_size, void* d_ws, size_t ws_size,
                              hipStream_t stream) {
    const float* x    = (const float*)d_in[0];   // [N_ENT, 64]
    const float* w    = (const float*)d_in[1];   // [41, 64, 64]
    const float* bias = (const float*)d_in[2];   // [64]
    const float* rvec = (const float*)d_in[3];   // [40, 64]
    const int*   ei   = (const int*)d_in[4];     // [2, N_EDGES]
    const int*   et   = (const int*)d_in[5];     // [N_EDGES]
    const int*   fr   = ei;
    const int*   to   = ei + N_EDGES;
    float* out = (float*)d_out;

    // Workspace layout: deg (u32, R*N) | Y (f32, R*N*64)
    unsigned* deg = (unsigned*)d_ws;
    size_t degBytes = (((size_t)NUM_REL * N_ENT * sizeof(unsigned)) + 255u) & ~(size_t)255u;
    float* y = (float*)((char*)d_ws + degBytes);

    rgcn_zero_kernel<<<2048, 256, 0, stream>>>(out, deg, N_ENT * DIM, NUM_REL * N_ENT);
    rgcn_degree_kernel<<<2048, 256, 0, stream>>>(fr, et, deg);

    dim3 ggrid((N_ENT + 31) / 32, NUM_REL);
    rgcn_gemm_wmma<<<ggrid, 256, 0, stream>>>(x, w, y);

    rgcn_scatter_kernel<<<N_EDGES / 16, 256, 0, stream>>>(fr, to, et, deg, y, out);
    rgcn_finalize_kernel<<<2048, 256, 0, stream>>>(y, bias, rvec, out);
}